// TemporalGATLayer_87677462381156
// MI455X (gfx1250) — compile-verified
//
#include <hip/hip_runtime.h>
#include <hip/hip_bf16.h>
#include <math.h>

#define IN_DIM   128
#define OUT_DIM  128
#define D_TIME   64
#define ATTN_LEN (2 * OUT_DIM + D_TIME)   // 320

typedef __attribute__((ext_vector_type(2))) float v2f;
typedef __attribute__((ext_vector_type(8))) float v8f;

// ---------- helpers: order-preserving float<->uint for atomicMax ----------
__device__ __forceinline__ unsigned flipf(float f) {
  unsigned u = __float_as_uint(f);
  return (u & 0x80000000u) ? ~u : (u | 0x80000000u);
}
__device__ __forceinline__ float unflipf(unsigned u) {
  return __uint_as_float((u & 0x80000000u) ? (u & 0x7FFFFFFFu) : ~u);
}
// flip(-inf) == 0x007FFFFF
#define FLIP_NEG_INF 0x007FFFFFu

// ---------- 1) u_s = W_src^T a_s ; u_d = W_dst^T a_d ----------
__global__ void prep_u_kernel(const float* __restrict__ W_src,
                              const float* __restrict__ W_dst,
                              const float* __restrict__ attn_w,
                              float* __restrict__ u_s,
                              float* __restrict__ u_d) {
  int k = threadIdx.x;              // 0..127
  float ss = 0.f, sd = 0.f;
  for (int o = 0; o < OUT_DIM; ++o) {
    ss += attn_w[o]           * W_src[o * IN_DIM + k];
    sd += attn_w[OUT_DIM + o] * W_dst[o * IN_DIM + k];
  }
  u_s[k] = ss;
  u_d[k] = sd;
}

// ---------- 2) zero out / seg_sum, init seg_max ----------
__global__ void init_kernel(float* __restrict__ out, int n_out,
                            float* __restrict__ seg_sum,
                            unsigned* __restrict__ seg_max, int n_seg) {
  int stride = gridDim.x * blockDim.x;
  for (int j = blockIdx.x * blockDim.x + threadIdx.x; j < n_out; j += stride)
    out[j] = 0.f;
  for (int j = blockIdx.x * blockDim.x + threadIdx.x; j < n_seg; j += stride) {
    seg_sum[j] = 0.f;
    seg_max[j] = FLIP_NEG_INF;
  }
}

// ---------- 3) Z_src = h_src @ W_src^T via V_WMMA_F32_16X16X4_F32 ----------
// Block: 256 threads = 8 waves; wave w owns output cols [16w, 16w+16).
// Grid.x: node tile (16 rows). K=128 -> 32 wmma steps of K=4.
__global__ void __launch_bounds__(256)
gemm_zsrc_kernel(const float* __restrict__ h,
                 const float* __restrict__ W,   // (OUT_DIM, IN_DIM) row-major
                 float* __restrict__ Z,         // (n_nodes, OUT_DIM)
                 int n_nodes) {
  __shared__ float Wlds[OUT_DIM * IN_DIM];      // 64 KB
  for (int i = threadIdx.x; i < OUT_DIM * IN_DIM; i += 256)
    Wlds[i] = W[i];
  __syncthreads();

  const int lane   = threadIdx.x & 31;
  const int waveId = threadIdx.x >> 5;
  const int n0     = waveId * 16;               // N tile base (output col)
  const int row0   = blockIdx.x * 16;           // M tile base (node row)
  if (row0 >= n_nodes) return;

  // A layout (16x4 f32): lanes 0-15 -> M=lane, {K0,K1}; lanes 16-31 -> M=lane-16, {K2,K3}
  // B layout (4x16 f32): lanes 0-15 -> N=lane,  {K0,K1}; lanes 16-31 -> N=lane-16, {K2,K3}
  const int mrow   = row0 + (lane & 15);
  const int ncol   = n0   + (lane & 15);
  const int kphase = (lane >> 4) * 2;           // 0 or 2

  const float* hrow = h    + (size_t)mrow * IN_DIM + kphase;
  const float* wrow = Wlds + ncol * IN_DIM + kphase;   // B[k][n] = W[n][k]

  v8f c = {0.f, 0.f, 0.f, 0.f, 0.f, 0.f, 0.f, 0.f};
#pragma unroll
  for (int k0 = 0; k0 < IN_DIM; k0 += 4) {
    v2f a = *(const v2f*)(hrow + k0);
    v2f b = *(const v2f*)(wrow + k0);
    c = __builtin_amdgcn_wmma_f32_16x16x4_f32(
        /*neg_a=*/false, a, /*neg_b=*/false, b,
        /*c_mod=*/(short)0, c, /*reuse_a=*/false, /*reuse_b=*/false);
  }

  // C layout: VGPR v, lanes 0-15 -> (M=v, N=lane); lanes 16-31 -> (M=v+8, N=lane-16)
  const int rowBase = row0 + (lane >> 4) * 8;
  float* zp = Z + (size_t)rowBase * OUT_DIM + ncol;
#pragma unroll
  for (int v = 0; v < 8; ++v)
    zp[(size_t)v * OUT_DIM] = c[v];
}

// ---------- 4) s_src[i] = h_src[i].u_s ; s_dst[j] = h_dst[j].u_d ----------
__global__ void node_scores_kernel(const float* __restrict__ h_src,
                                   const float* __restrict__ h_dst,
                                   const float* __restrict__ u_s,
                                   const float* __restrict__ u_d,
                                   float* __restrict__ s_src,
                                   float* __restrict__ s_dst,
                                   int n_src, int n_dst) {
  int wave = (blockIdx.x * blockDim.x + threadIdx.x) >> 5;
  int lane = threadIdx.x & 31;
  if (wave >= n_src + n_dst) return;
  const float* h; const float* u; float* s; int node;
  if (wave < n_src) { h = h_src; u = u_s; s = s_src; node = wave; }
  else              { h = h_dst; u = u_d; s = s_dst; node = wave - n_src; }

  const float4* hp = (const float4*)(h + (size_t)node * IN_DIM);
  const float4* up = (const float4*)u;
  float4 hv = hp[lane];
  float4 uv = up[lane];
  float acc = hv.x * uv.x + hv.y * uv.y + hv.z * uv.z + hv.w * uv.w;
#pragma unroll
  for (int off = 16; off > 0; off >>= 1)
    acc += __shfl_down(acc, off, 32);
  if (lane == 0) s[node] = acc;
}

// ---------- 5) per-edge score + segment max (wave per edge) ----------
__global__ void edge_scores_kernel(const int* __restrict__ edge_index,
                                   const float* __restrict__ phi,
                                   const float* __restrict__ attn_w,
                                   const float* __restrict__ s_src,
                                   const float* __restrict__ s_dst,
                                   float* __restrict__ e_buf,
                                   unsigned* __restrict__ seg_max,
                                   int nE, float inv_scale) {
  int edge = (blockIdx.x * blockDim.x + threadIdx.x) >> 5;
  int lane = threadIdx.x & 31;
  if (edge >= nE) return;

  const float2* pp = (const float2*)(phi + (size_t)edge * D_TIME);
  float2 p  = pp[lane];
  float at0 = attn_w[2 * OUT_DIM + 2 * lane];
  float at1 = attn_w[2 * OUT_DIM + 2 * lane + 1];
  float acc = p.x * at0 + p.y * at1;
#pragma unroll
  for (int off = 16; off > 0; off >>= 1)
    acc += __shfl_down(acc, off, 32);

  if (lane == 0) {
    int s = edge_index[edge];
    int d = edge_index[nE + edge];
    float v = acc + s_src[s] + s_dst[d];
    v = (v > 0.f ? v : 0.2f * v) * inv_scale;   // leaky_relu(0.2)/SCALE
    e_buf[edge] = v;
    atomicMax(seg_max + d, flipf(v));
  }
}

// ---------- 6) ex = exp(e - seg_max[dst]); seg_sum += ex ----------
__global__ void edge_exp_kernel(const int* __restrict__ edge_index,
                                float* __restrict__ e_buf,
                                const unsigned* __restrict__ seg_max,
                                float* __restrict__ seg_sum, int nE) {
  int k = blockIdx.x * blockDim.x + threadIdx.x;
  if (k >= nE) return;
  int d = edge_index[nE + k];
  float m  = unflipf(seg_max[d]);
  float ex = __expf(e_buf[k] - m);
  e_buf[k] = ex;
  atomicAdd(seg_sum + d, ex);
}

// ---------- 7) out[dst] += alpha * Z_src[src]  (wave per edge) ----------
__global__ void scatter_kernel(const int* __restrict__ edge_index,
                               const float* __restrict__ e_buf,
                               const float* __restrict__ seg_sum,
                               const float* __restrict__ Z,
                               float* __restrict__ out, int nE) {
  int edge = (blockIdx.x * blockDim.x + threadIdx.x) >> 5;
  int lane = threadIdx.x & 31;
  if (edge >= nE) return;

  int s = edge_index[edge];
  int d = edge_index[nE + edge];
  float alpha = e_buf[edge] / (seg_sum[d] + 1e-12f);

  const float4* zp = (const float4*)(Z + (size_t)s * OUT_DIM);
  float4 z = zp[lane];
  float* o = out + (size_t)d * OUT_DIM + lane * 4;
  atomicAdd(o + 0, alpha * z.x);
  atomicAdd(o + 1, alpha * z.y);
  atomicAdd(o + 2, alpha * z.z);
  atomicAdd(o + 3, alpha * z.w);
}

extern "C" void kernel_launch(void* const* d_in, const int* in_sizes, int n_in,
                              void* d_out, int out_size, void* d_ws, size_t ws_size,
                              hipStream_t stream) {
  const float* h_src  = (const float*)d_in[0];
  const float* h_dst  = (const float*)d_in[1];
  const int*   eidx   = (const int*)  d_in[2];
  const float* phi_dt = (const float*)d_in[3];
  const float* W_src  = (const float*)d_in[4];
  const float* W_dst  = (const float*)d_in[5];
  const float* attn_w = (const float*)d_in[6];
  float* out = (float*)d_out;

  const int n_src = in_sizes[0] / IN_DIM;     // 50000
  const int n_dst = in_sizes[1] / IN_DIM;     // 50000
  const int nE    = in_sizes[2] / 2;          // 800000
  const float inv_scale = 1.0f / sqrtf((float)ATTN_LEN);

  // workspace layout (bytes)
  char* ws = (char*)d_ws;
  size_t off = 0;
  float*    Z       = (float*)(ws + off); off += (size_t)n_src * OUT_DIM * 4;
  float*    e_buf   = (float*)(ws + off); off += (size_t)nE * 4;
  float*    s_src   = (float*)(ws + off); off += (size_t)n_src * 4;
  float*    s_dst   = (float*)(ws + off); off += (size_t)n_dst * 4;
  float*    seg_sum = (float*)(ws + off); off += (size_t)n_dst * 4;
  unsigned* seg_max = (unsigned*)(ws + off); off += (size_t)n_dst * 4;
  float*    u_s     = (float*)(ws + off); off += 512;
  float*    u_d     = (float*)(ws + off); off += 512;
  (void)ws_size; (void)n_in;

  // 1) fold attn weights into projections
  prep_u_kernel<<<1, 128, 0, stream>>>(W_src, W_dst, attn_w, u_s, u_d);

  // 2) init accumulators
  init_kernel<<<2048, 256, 0, stream>>>(out, out_size, seg_sum, seg_max, n_dst);

  // 3) Z_src = h_src @ W_src^T  (WMMA fp32)
  gemm_zsrc_kernel<<<(n_src + 15) / 16, 256, 0, stream>>>(h_src, W_src, Z, n_src);

  // 4) per-node attention scalars
  {
    int waves = n_src + n_dst;
    int blocks = (waves * 32 + 255) / 256;
    node_scores_kernel<<<blocks, 256, 0, stream>>>(h_src, h_dst, u_s, u_d,
                                                   s_src, s_dst, n_src, n_dst);
  }

  // 5) edge scores + segment max
  {
    int blocks = (nE * 32 + 255) / 256;   // 8 waves per block
    edge_scores_kernel<<<blocks, 256, 0, stream>>>(eidx, phi_dt, attn_w,
                                                   s_src, s_dst, e_buf, seg_max,
                                                   nE, inv_scale);
  }

  // 6) exp + segment sum
  edge_exp_kernel<<<(nE + 255) / 256, 256, 0, stream>>>(eidx, e_buf, seg_max,
                                                        seg_sum, nE);

  // 7) weighted scatter of messages
  {
    int blocks = (nE * 32 + 255) / 256;
    scatter_kernel<<<blocks, 256, 0, stream>>>(eidx, e_buf, seg_sum, Z, out, nE);
  }
}